// GCN_22874995819126
// MI455X (gfx1250) — compile-verified
//
#include <hip/hip_runtime.h>
#include <hip/hip_bf16.h>

typedef float v2f __attribute__((ext_vector_type(2)));
typedef float v8f __attribute__((ext_vector_type(8)));

#define NNODES 100000
#define NEDGES 1200000
#define NGRAPH 64

// ---------------- degree / normalization ----------------
__global__ void k_deg_init(float* deg, int n) {
    int i = blockIdx.x * blockDim.x + threadIdx.x;
    if (i < n) deg[i] = 1.0f;                       // self-loop
}

__global__ void k_deg_count(const int* __restrict__ dst, float* deg, int e) {
    int i = blockIdx.x * blockDim.x + threadIdx.x;
    if (i < e) atomicAdd(&deg[dst[i]], 1.0f);
}

__global__ void k_rsqrt_inplace(float* d, int n) {
    int i = blockIdx.x * blockDim.x + threadIdx.x;
    if (i < n) d[i] = rsqrtf(d[i]);
}

// ---------------- WMMA fp32 linear: out[N x 64] = h[N x K] @ W[K x 64] ----------------
// One wave -> one 16-row x 64-col output tile (four 16x16 accumulators).
// A frag (16x4 f32): lanes 0-15 supply K=k0,k0+1 for row M=lane; lanes 16-31 supply K=k0+2,k0+3.
// B frag (4x16 f32): same per-half K split, col N = lane%16.
// D (16x16 f32): VGPR j -> row = j + 8*(lane/16), col = lane%16.
template <int K>
__global__ void k_linear_wmma(const float* __restrict__ h, const float* __restrict__ W,
                              float* __restrict__ out, int nrows) {
    int wave = (blockIdx.x * blockDim.x + threadIdx.x) >> 5;
    int lane = threadIdx.x & 31;
    int ntiles = nrows >> 4;                         // nrows % 16 == 0
    if (wave >= ntiles) return;                      // wave-uniform: EXEC stays all-ones
    int half = lane >> 4;
    int m    = lane & 15;
    int row  = (wave << 4) + m;
    const float* hrow = h + (size_t)row * K;

    v8f c0 = {}, c1 = {}, c2 = {}, c3 = {};
    for (int k0 = 0; k0 < K; k0 += 4) {
        int ka = k0 + 2 * half;
        v2f a;
        a.x = hrow[ka];
        a.y = hrow[ka + 1];
        const float* w0 = W + (size_t)ka * 64;       // row ka, row ka+1 is +64
        v2f b;
        b.x = w0[m];            b.y = w0[64 + m];
        c0 = __builtin_amdgcn_wmma_f32_16x16x4_f32(false, a, false, b, (short)0, c0, false, false);
        b.x = w0[16 + m];       b.y = w0[64 + 16 + m];
        c1 = __builtin_amdgcn_wmma_f32_16x16x4_f32(false, a, false, b, (short)0, c1, false, false);
        b.x = w0[32 + m];       b.y = w0[64 + 32 + m];
        c2 = __builtin_amdgcn_wmma_f32_16x16x4_f32(false, a, false, b, (short)0, c2, false, false);
        b.x = w0[48 + m];       b.y = w0[64 + 48 + m];
        c3 = __builtin_amdgcn_wmma_f32_16x16x4_f32(false, a, false, b, (short)0, c3, false, false);
    }
    #pragma unroll
    for (int j = 0; j < 8; ++j) {
        size_t ro = ((size_t)((wave << 4) + half * 8 + j)) * 64;
        out[ro +  0 + m] = c0[j];
        out[ro + 16 + m] = c1[j];
        out[ro + 32 + m] = c2[j];
        out[ro + 48 + m] = c3[j];
    }
}

// ---------------- self-loop term + bias (initializes agg buffer) ----------------
__global__ void k_self_bias(const float* __restrict__ hW, const float* __restrict__ dinv,
                            const float* __restrict__ bias, float* __restrict__ out, int n) {
    int idx = blockIdx.x * blockDim.x + threadIdx.x;
    if (idx >= n * 64) return;
    int i = idx >> 6, c = idx & 63;
    float d = dinv[i];
    out[idx] = hW[idx] * (d * d) + bias[c];
}

// ---------------- edge scatter-add: out[dst] += hW[src] * dinv[src]*dinv[dst] ----------------
// 16 threads per edge, 4 channels each (float4 gather, 4 f32 atomics).
__global__ void k_edge_scatter(const int* __restrict__ src, const int* __restrict__ dst,
                               const float* __restrict__ dinv, const float* __restrict__ hW,
                               float* __restrict__ out, int e) {
    int tid = blockIdx.x * blockDim.x + threadIdx.x;
    int edge = tid >> 4;
    if (edge >= e) return;
    int part = (tid & 15) << 2;
    int s = src[edge], d = dst[edge];
    float w = dinv[s] * dinv[d];
    const float4 v = *(const float4*)(hW + (size_t)s * 64 + part);
    float* o = out + (size_t)d * 64 + part;
    atomicAdd(o + 0, v.x * w);
    atomicAdd(o + 1, v.y * w);
    atomicAdd(o + 2, v.z * w);
    atomicAdd(o + 3, v.w * w);
}

// ---------------- fused BatchNorm(eval) + ReLU, in place ----------------
__global__ void k_bnrelu(float* __restrict__ h, const float* __restrict__ gamma,
                         const float* __restrict__ beta, const float* __restrict__ mean,
                         const float* __restrict__ var, int n) {
    int idx = blockIdx.x * blockDim.x + threadIdx.x;
    if (idx >= n * 64) return;
    int c = idx & 63;
    float v = (h[idx] - mean[c]) * rsqrtf(var[c] + 1e-5f) * gamma[c] + beta[c];
    h[idx] = v > 0.0f ? v : 0.0f;
}

// ---------------- pooling ----------------
__global__ void k_zero(float* p, int n) {
    int i = blockIdx.x * blockDim.x + threadIdx.x;
    if (i < n) p[i] = 0.0f;
}

__global__ void k_pool_acc(const float* __restrict__ h, const int* __restrict__ batch,
                           float* __restrict__ pooled, float* __restrict__ cnt, int n) {
    int tid = blockIdx.x * blockDim.x + threadIdx.x;
    int i = tid >> 4;
    if (i >= n) return;
    int part = (tid & 15) << 2;
    int g = batch[i];
    const float4 v = *(const float4*)(h + (size_t)i * 64 + part);
    float* o = pooled + (size_t)g * 64 + part;
    atomicAdd(o + 0, v.x);
    atomicAdd(o + 1, v.y);
    atomicAdd(o + 2, v.z);
    atomicAdd(o + 3, v.w);
    if ((tid & 15) == 0) atomicAdd(&cnt[g], 1.0f);
}

// ---------------- MLP head (single block): mean-div, lin1+relu, lin2 ----------------
__global__ void k_head(const float* __restrict__ pooled, const float* __restrict__ cnt,
                       const float* __restrict__ w1, const float* __restrict__ b1,
                       const float* __restrict__ w2, const float* __restrict__ b2,
                       float* __restrict__ out) {
    __shared__ float sp[64 * 64];
    __shared__ float sh[64 * 32];
    int t = threadIdx.x;
    for (int idx = t; idx < 64 * 64; idx += 256) {
        int g = idx >> 6;
        float c = cnt[g];
        if (c < 1.0f) c = 1.0f;
        sp[idx] = pooled[idx] / c;
    }
    __syncthreads();
    for (int idx = t; idx < 64 * 32; idx += 256) {
        int g = idx >> 5, o = idx & 31;
        float acc = b1[o];
        for (int k = 0; k < 64; ++k) acc += sp[g * 64 + k] * w1[k * 32 + o];
        sh[idx] = acc > 0.0f ? acc : 0.0f;
    }
    __syncthreads();
    for (int idx = t; idx < 64 * 8; idx += 256) {
        int g = idx >> 3, o = idx & 7;
        float acc = b2[o];
        for (int k = 0; k < 32; ++k) acc += sh[g * 32 + k] * w2[k * 8 + o];
        out[idx] = acc;
    }
}

extern "C" void kernel_launch(void* const* d_in, const int* in_sizes, int n_in,
                              void* d_out, int out_size, void* d_ws, size_t ws_size,
                              hipStream_t stream) {
    (void)in_sizes; (void)n_in; (void)out_size; (void)ws_size;
    const int N = NNODES, E = NEDGES, G = NGRAPH;

    const float* x     = (const float*)d_in[0];
    const int*   ei    = (const int*)d_in[1];
    const int*   src   = ei;
    const int*   dst   = ei + E;
    const int*   batch = (const int*)d_in[2];

    const float *W[4], *b[4], *ga[4], *be[4], *mn[4], *vr[4];
    for (int l = 0; l < 4; ++l) {
        W[l]  = (const float*)d_in[3 + 6 * l];
        b[l]  = (const float*)d_in[4 + 6 * l];
        ga[l] = (const float*)d_in[5 + 6 * l];
        be[l] = (const float*)d_in[6 + 6 * l];
        mn[l] = (const float*)d_in[7 + 6 * l];
        vr[l] = (const float*)d_in[8 + 6 * l];
    }
    const float* w1 = (const float*)d_in[27];
    const float* b1 = (const float*)d_in[28];
    const float* w2 = (const float*)d_in[29];
    const float* b2 = (const float*)d_in[30];
    float* out = (float*)d_out;

    // workspace layout (floats): dinv[N] | buf1[64N] | buf2[64N] | pooled[G*64] | cnt[G]
    float* ws     = (float*)d_ws;
    float* dinv   = ws;
    float* buf1   = ws + N;
    float* buf2   = buf1 + (size_t)N * 64;
    float* pooled = buf2 + (size_t)N * 64;
    float* cnt    = pooled + (size_t)G * 64;

    // degrees -> deg^{-1/2}
    k_deg_init<<<(N + 255) / 256, 256, 0, stream>>>(dinv, N);
    k_deg_count<<<(E + 255) / 256, 256, 0, stream>>>(dst, dinv, E);
    k_rsqrt_inplace<<<(N + 255) / 256, 256, 0, stream>>>(dinv, N);

    const int tiles     = N / 16;                 // 6250, exact
    const int linBlocks = (tiles + 7) / 8;        // 8 waves (256 thr) per block
    const int ewBlocks  = (N * 64 + 255) / 256;
    const int esBlocks  = (E * 16 + 255) / 256;

    for (int l = 0; l < 4; ++l) {
        const float* hin = (l == 0) ? x : buf2;
        if (l == 0)
            k_linear_wmma<32><<<linBlocks, 256, 0, stream>>>(hin, W[l], buf1, N);
        else
            k_linear_wmma<64><<<linBlocks, 256, 0, stream>>>(hin, W[l], buf1, N);
        k_self_bias<<<ewBlocks, 256, 0, stream>>>(buf1, dinv, b[l], buf2, N);
        k_edge_scatter<<<esBlocks, 256, 0, stream>>>(src, dst, dinv, buf1, buf2, E);
        k_bnrelu<<<ewBlocks, 256, 0, stream>>>(buf2, ga[l], be[l], mn[l], vr[l], N);
    }

    k_zero<<<(G * 64 + G + 255) / 256, 256, 0, stream>>>(pooled, G * 64 + G); // pooled+cnt contiguous
    k_pool_acc<<<(N * 16 + 255) / 256, 256, 0, stream>>>(buf2, batch, pooled, cnt, N);
    k_head<<<1, 256, 0, stream>>>(pooled, cnt, w1, b1, w2, b2, out);
}